// SpecAugment_4801773436929
// MI455X (gfx1250) — compile-verified
//
#include <hip/hip_runtime.h>
#include <stdint.h>

// SpecAugment: x[B=64, C=1, F=128, T=3000] fp32; zero freq band [f0,f0+f) and
// time band [t0,t0+t) per batch. Pure streaming op (~196.6 MB, 0 FLOPs);
// roofline at 23.3 TB/s is ~8.4 us.
//
// CDNA5 implementation strategy:
//  - one block (8 wave32) per (b, freq) row; 8192 blocks
//  - bulk bytes ride the gfx1250 async copy path:
//      global_load_async_to_lds_b128  (ASYNCcnt)  global -> LDS
//      ds_store of <=30 zeros for the time band
//      global_store_async_from_lds_b128 th:TH_STORE_NT  LDS -> global
//    so row data never stages through VGPRs
//  - freq-masked rows skip the read entirely: NT zero stores straight out
//  - NT store hint keeps the 98.3 MB input resident in the 192 MB L2 across
//    graph replays (input+output = 196.6 MB would not both fit)

typedef float v4f __attribute__((ext_vector_type(4)));

namespace {
constexpr int kB = 64;
constexpr int kF = 128;
constexpr int kT = 3000;
constexpr int kT4 = kT / 4;        // 750 b128 transfers per row
constexpr int kThreads = 256;      // 8 wave32 per block
constexpr int kRows = kB * kF;     // 8192 blocks
static_assert(kT % 4 == 0, "row must vectorize by 4 floats");
}  // namespace

__global__ __launch_bounds__(kThreads) void specaugment_async_kernel(
    const float* __restrict__ x,
    const int* __restrict__ t,  const int* __restrict__ f,
    const int* __restrict__ t0, const int* __restrict__ f0,
    float* __restrict__ out)
{
    __shared__ float row[kT];      // 12 KB -> ~26 blocks resident per WGP

    const int r   = (int)blockIdx.x;   // row id = b*kF + fi
    const int b   = r >> 7;            // / kF
    const int fi  = r & (kF - 1);
    const int tid = (int)threadIdx.x;

    const int tlen = t[b];
    const int flen = f[b];
    const int tlo  = t0[b];
    const int flo  = f0[b];

    const long rowOff = (long)r * kT;
    float* __restrict__ outRow = out + rowOff;

    if ((unsigned)(fi - flo) < (unsigned)flen) {
        // Whole row zero: skip the global read AND the LDS round trip.
        v4f z = (v4f)0.0f;
        v4f* __restrict__ o4 = (v4f*)outRow;
        for (int i = tid; i < kT4; i += kThreads)
            __builtin_nontemporal_store(z, &o4[i]);
        return;
    }

    const float* __restrict__ xRow = x + rowOff;
    // LDS byte address = low 32 bits of the generic shared pointer
    // (flat LDS aperture truncates to addr[31:0]).
    const unsigned ldsBase = (unsigned)(uintptr_t)(void*)row;

    // ---- async bulk copy: global -> LDS (3 b128 issues per thread) ----
    for (int i = tid; i < kT4; i += kThreads) {
        const unsigned goff = (unsigned)(i * 16);       // byte offset in row
        const unsigned loff = ldsBase + (unsigned)(i * 16);
        asm volatile("global_load_async_to_lds_b128 %0, %1, %2"
                     :
                     : "v"(loff), "v"(goff), "s"(xRow)
                     : "memory");
    }
    asm volatile("s_wait_asynccnt 0x0" ::: "memory");   // my loads landed
    __syncthreads();                                    // everyone's landed

    // ---- zero the time band (width <= 30, first wave only) ----
    if (tid < tlen) row[tlo + tid] = 0.0f;
    __syncthreads();                                    // zeros visible

    // ---- async bulk copy: LDS -> global, non-temporal ----
    for (int i = tid; i < kT4; i += kThreads) {
        const unsigned goff = (unsigned)(i * 16);
        const unsigned loff = ldsBase + (unsigned)(i * 16);
        asm volatile("global_store_async_from_lds_b128 %0, %1, %2 th:TH_STORE_NT"
                     :
                     : "v"(goff), "v"(loff), "s"(outRow)
                     : "memory");
    }
    asm volatile("s_wait_asynccnt 0x0" ::: "memory");
}

extern "C" void kernel_launch(void* const* d_in, const int* in_sizes, int n_in,
                              void* d_out, int out_size, void* d_ws, size_t ws_size,
                              hipStream_t stream)
{
    (void)in_sizes; (void)n_in; (void)out_size; (void)d_ws; (void)ws_size;
    const float* x  = (const float*)d_in[0];
    const int*   t  = (const int*)d_in[1];
    const int*   f  = (const int*)d_in[2];
    const int*   t0 = (const int*)d_in[3];
    const int*   f0 = (const int*)d_in[4];
    float* out = (float*)d_out;

    specaugment_async_kernel<<<kRows, kThreads, 0, stream>>>(x, t, f, t0, f0, out);
}